// LossWithErrorRegularization_87969520157002
// MI455X (gfx1250) — compile-verified
//
#include <hip/hip_runtime.h>

#define N_ROWS    8192
#define N_CLASSES 1000
#define LAMBDA_P  0.01f

typedef __attribute__((ext_vector_type(2))) float v2f;
typedef __attribute__((ext_vector_type(8))) float v8f;

#if __has_builtin(__builtin_amdgcn_tensor_load_to_lds)
#define HAVE_TDM 1
typedef unsigned int v4u __attribute__((ext_vector_type(4)));
typedef int          v4i __attribute__((ext_vector_type(4)));
typedef int          v8i __attribute__((ext_vector_type(8)));
#else
#define HAVE_TDM 0
#endif

// ws layout (floats):
//   [0] = ce accumulator, [1] = penalty accumulator
//   [8 .. 8+2N)      : Apair[i] = {a_i*c_i, a_i}          (float2, interleaved)
//   [8+2N .. 8+4N)   : Bpair[j] = {b_j, -b_j*c_j}         (float2, interleaved)

__global__ void init_accum_kernel(float* __restrict__ ws) {
    if (threadIdx.x == 0) { ws[0] = 0.0f; ws[1] = 0.0f; }
}

// One wave32 per row: max/argmax + log-sum-exp + CE gather, and emit the
// rank-2 factor pairs for the WMMA penalty kernel.
__global__ void ce_argmax_kernel(const float* __restrict__ outputs,
                                 const float* __restrict__ conf,
                                 const int*   __restrict__ labels,
                                 float* __restrict__ ws) {
    __shared__ float blockCE;
    const int lane = threadIdx.x & 31;
    const int wave = threadIdx.x >> 5;
    const int row  = blockIdx.x * 8 + wave;
    if (threadIdx.x == 0) blockCE = 0.0f;
    __syncthreads();

    const float* __restrict__ rowp = outputs + (size_t)row * N_CLASSES;
    __builtin_prefetch(rowp + lane, 0, 1);   // gfx1250 global_prefetch_b8

    // pass 1: max + argmax (first-max tie-break like jnp.argmax)
    float maxV = -3.402823466e38f;
    int   maxI = 0x7fffffff;
    for (int c = lane; c < N_CLASSES; c += 32) {
        float v = rowp[c];
        if (v > maxV || (v == maxV && c < maxI)) { maxV = v; maxI = c; }
    }
    for (int off = 16; off > 0; off >>= 1) {
        float ov = __shfl_xor(maxV, off, 32);
        int   oi = __shfl_xor(maxI, off, 32);
        if (ov > maxV || (ov == maxV && oi < maxI)) { maxV = ov; maxI = oi; }
    }

    // pass 2: sum of exp(x - max); tensor (32MB) is resident in the 192MB L2
    float s = 0.0f;
    for (int c = lane; c < N_CLASSES; c += 32) s += __expf(rowp[c] - maxV);
    for (int off = 16; off > 0; off >>= 1) s += __shfl_xor(s, off, 32);

    if (lane == 0) {
        const int   lbl = labels[row];
        const float lp  = rowp[lbl] - maxV - __logf(s);  // log-softmax at label
        atomicAdd(&blockCE, lp);

        const float c     = conf[row];
        const float truth = (maxI == lbl) ? 1.0f : 0.0f;
        const float wrong = 1.0f - truth;
        float2* __restrict__ Apair = (float2*)(ws + 8);
        float2* __restrict__ Bpair = Apair + N_ROWS;
        Apair[row] = make_float2(wrong * c, wrong);     // {a_i*c_i, a_i}
        Bpair[row] = make_float2(truth, -truth * c);    // {b_j, -b_j*c_j}
    }
    __syncthreads();
    if (threadIdx.x == 0) atomicAdd(&ws[0], blockCE);
}

// One wave32 per (i-tile, j-chunk). D = A x B via V_WMMA_F32_16X16X4_F32 gives
// the masked pairwise-diff tile a_i*b_j*(c_i - c_j); VALU does relu^2 + accum.
// B rows K=2,3 are don't-care because A's K=2,3 columns are zero, so every
// lane loads B unconditionally (no EXEC predication in the hot loop).
__global__ void penalty_wmma_kernel(float* __restrict__ ws) {
    const float2* __restrict__ Apair = (const float2*)(ws + 8);
    const float2* __restrict__ Bpair = Apair + N_ROWS;

    const int lane   = threadIdx.x;       // blockDim.x == 32
    const int iTile  = blockIdx.x;        // 0..511
    const int jChunk = blockIdx.y;        // 0..7, 64 j-tiles each
    const int jl     = lane & 15;
    const int j0     = jChunk * 64;

    // A (16x4 f32, 2 VGPRs): lanes 0-15 carry rows M=lane with K=0,1;
    // lanes 16-31 carry K=2,3 which stay zero (makes B rows 2,3 don't-care).
    float a0 = 0.0f, a1 = 0.0f;
    if (lane < 16) {
        const float2 aa = Apair[iTile * 16 + lane];
        a0 = aa.x;
        a1 = aa.y;
    }
    const v2f Av = {a0, a1};

    float acc = 0.0f;

#if HAVE_TDM
    // TDM: DMA this block's whole B chunk (64 tiles * 16 rows * 8B = 8KB)
    // into LDS once, then feed the WMMA loop from ds_load_b64.
    __shared__ float2 ldsB[64 * 16];

    const unsigned      ldsOff  = (unsigned)(size_t)
        (__attribute__((address_space(3))) void*)&ldsB[0];
    const unsigned long gaddr   = (unsigned long)(const void*)(Bpair + j0 * 16);
    const unsigned      nElem8B = 64u * 16u;   // 1024 8-byte elements

    // D# group 0: count=1 | lds_addr | global_addr[56:0] | type=2
    v4u g0;
    g0.x = 1u;                                     // count=1, user mode
    g0.y = ldsOff;                                 // lds_addr (bytes)
    g0.z = (unsigned)(gaddr & 0xffffffffu);        // global_addr[31:0]
    g0.w = (unsigned)((gaddr >> 32) & 0x01ffffffu) // global_addr[56:32]
         | (2u << 30);                             // type=2 ("image")
    // D# group 1: 1-D tile of 1024 x 8-byte elements
    v8i g1;
    g1[0] = (int)(3u << 16);          // workgroup_mask=0, data_size=3 (8B)
    g1[1] = (int)(nElem8B << 16);     // tensor_dim0[15:0] in bits 79:64... low half
    g1[2] = (int)(1u << 16);          // tensor_dim0 hi16=0 | tensor_dim1=1 lo16
    g1[3] = (int)(nElem8B << 16);     // tensor_dim1 hi16=0 | tile_dim0=1024
    g1[4] = 1;                        // tile_dim1=1 | tile_dim2=0
    g1[5] = (int)nElem8B;             // tensor_dim0_stride[31:0]
    g1[6] = 0;                        // stride hi | tensor_dim1_stride lo
    g1[7] = 0;
    v4i z4 = {0, 0, 0, 0};
#if __clang_major__ >= 23
    v8i z8 = {0, 0, 0, 0, 0, 0, 0, 0};
    __builtin_amdgcn_tensor_load_to_lds(g0, g1, z4, z4, z8, 0);
#else
    __builtin_amdgcn_tensor_load_to_lds(g0, g1, z4, z4, 0);
#endif
    __builtin_amdgcn_s_wait_tensorcnt(0);   // single wave per WG: no barrier

    for (int t = 0; t < 64; ++t) {
        const float2 bb = ldsB[t * 16 + jl];        // ds_load_b64, all lanes
        const v2f Bv = {bb.x, bb.y};
        v8f Cz = {0.0f, 0.0f, 0.0f, 0.0f, 0.0f, 0.0f, 0.0f, 0.0f};
        v8f D = __builtin_amdgcn_wmma_f32_16x16x4_f32(
            false, Av, false, Bv, (short)0, Cz, false, false);
#pragma unroll
        for (int k = 0; k < 8; ++k) {
            const float r = fmaxf(D[k], 0.0f);
            acc = fmaf(r, r, acc);
        }
    }
#else
    for (int jt = j0; jt < j0 + 64; ++jt) {
        const float2 bb = Bpair[jt * 16 + jl];      // global_load_b64, all lanes
        const v2f Bv = {bb.x, bb.y};
        v8f Cz = {0.0f, 0.0f, 0.0f, 0.0f, 0.0f, 0.0f, 0.0f, 0.0f};
        v8f D = __builtin_amdgcn_wmma_f32_16x16x4_f32(
            false, Av, false, Bv, (short)0, Cz, false, false);
#pragma unroll
        for (int k = 0; k < 8; ++k) {
            const float r = fmaxf(D[k], 0.0f);
            acc = fmaf(r, r, acc);
        }
    }
#endif

    for (int off = 16; off > 0; off >>= 1) acc += __shfl_xor(acc, off, 32);
    if (lane == 0) atomicAdd(&ws[1], acc);
}

__global__ void combine_kernel(const float* __restrict__ ws,
                               float* __restrict__ out) {
    if (threadIdx.x == 0) {
        out[0] = -ws[0] / (float)N_ROWS + LAMBDA_P * ws[1];
    }
}

extern "C" void kernel_launch(void* const* d_in, const int* in_sizes, int n_in,
                              void* d_out, int out_size, void* d_ws, size_t ws_size,
                              hipStream_t stream) {
    const float* outputs = (const float*)d_in[0];   // [N, C] f32
    const float* conf    = (const float*)d_in[1];   // [N]    f32
    const int*   labels  = (const int*)  d_in[2];   // [N]    int
    float* ws  = (float*)d_ws;
    float* out = (float*)d_out;

    init_accum_kernel<<<1, 32, 0, stream>>>(ws);
    ce_argmax_kernel<<<N_ROWS / 8, 256, 0, stream>>>(outputs, conf, labels, ws);
    penalty_wmma_kernel<<<dim3(N_ROWS / 16, 8), 32, 0, stream>>>(ws);
    combine_kernel<<<1, 32, 0, stream>>>(ws, out);
}